// NLWT_21560735826113
// MI455X (gfx1250) — compile-verified
//
#include <hip/hip_runtime.h>

typedef __attribute__((ext_vector_type(2))) float v2f;
typedef __attribute__((ext_vector_type(8))) float v8f;

// Select M2[m][s] for m in 0..3, zero for m>=4 (A-matrix rows 4..15 are zero).
#define SEL4(m,c0,c1,c2,c3) \
  ((m)==0?(c0):((m)==1?(c1):((m)==2?(c2):((m)==3?(c3):0.0f))))

// x:   256 planes of 512x512 fp32  (b*c = 4*64 planes)
// out: 4 arrays, each 256 planes of 256x256 fp32, concatenated
__global__ __launch_bounds__(256) void nlwt_wmma_kernel(const float* __restrict__ x,
                                                        float* __restrict__ out) {
  const int lane = threadIdx.x & 31;   // wave32
  const int m    = lane & 15;          // output pixel within 16-wide chunk / A row
  const int half = lane >> 4;          // K-half selector for A/B layouts

  // A_s[m,l] = M2[m][s] * M1[s][l]  (rank-1), packed per the 16x4 f32 A layout:
  // lanes 0-15: VGPR0=K0, VGPR1=K1 ; lanes 16-31: VGPR0=K2, VGPR1=K3.
  const float w0 = SEL4(m, 1.3066f, -0.5412f, -0.4671f,  1.3349f); // M2[:,0]
  const float w1 = SEL4(m, 0.5412f,  1.3066f, -1.3349f, -0.4671f); // M2[:,1]
  const float w2 = SEL4(m, 0.5412f,  1.3066f,  1.3349f,  0.4671f); // M2[:,2]
  const float w3 = SEL4(m, 1.3066f, -0.5412f,  0.4671f, -1.3349f); // M2[:,3]

  v2f a0, a1, a2, a3;
  a0.x = w0 * (half ?  0.4083f :  0.9124f); a0.y = w0 * (half ?  0.0115f : -0.0256f); // M1[0,:]
  a1.x = w1 * (half ?  0.0115f :  0.0256f); a1.y = w1 * (half ? -0.4083f :  0.9124f); // M1[1,:]
  a2.x = w2 * (half ?  0.9124f : -0.4083f); a2.y = w2 * (half ?  0.0256f :  0.0115f); // M1[2,:]
  a3.x = w3 * (half ? -0.0256f :  0.0115f); a3.y = w3 * (half ?  0.9124f :  0.4083f); // M1[3,:]

  const int waveId     = blockIdx.x * (blockDim.x >> 5) + (threadIdx.x >> 5);
  const int wavesTotal = gridDim.x * (blockDim.x >> 5);
  const int nChunks    = 1 << 20;  // 256 planes * 256 output rows * 16 chunks/row

  for (int t = waveId; t < nChunks; t += wavesTotal) {  // uniform per wave
    const int qc    = t & 15;
    const int p     = (t >> 4) & 255;
    const int plane = t >> 12;
    const float* __restrict__ base = x + (size_t)plane * 262144u;

    const int q  = (qc << 4) + m;            // output column handled by this lane
    const int r0 = (2 * p     + half) & 511; // rows 2p / 2p+1   (wrap for roll)
    const int r1 = (2 * p + 2 + half) & 511; // rows 2p+2 / 2p+3 (wrap for roll)
    const int c0 = (2 * q    ) & 511;        // cols 2q, 2q+1
    const int c1 = (2 * q + 2) & 511;        // cols 2q+2, 2q+3  (wrap for roll)

    // B matrices (4x16 f32): one aligned float2 load per lane, coalesced
    // across the half-wave (lanes 0-15 row r even, lanes 16-31 row r odd).
    const v2f b00 = *(const v2f*)(base + (size_t)r0 * 512 + c0); // shift (0,0)
    const v2f b10 = *(const v2f*)(base + (size_t)r1 * 512 + c0); // shift (1,0)
    const v2f b01 = *(const v2f*)(base + (size_t)r0 * 512 + c1); // shift (0,1)
    const v2f b11 = *(const v2f*)(base + (size_t)r1 * 512 + c1); // shift (1,1)

    v8f d = {};
    d = __builtin_amdgcn_wmma_f32_16x16x4_f32(false, a0, false, b00, (short)0, d, false, false);
    d = __builtin_amdgcn_wmma_f32_16x16x4_f32(false, a1, false, b10, (short)0, d, false, false);
    d = __builtin_amdgcn_wmma_f32_16x16x4_f32(false, a2, false, b01, (short)0, d, false, false);
    d = __builtin_amdgcn_wmma_f32_16x16x4_f32(false, a3, false, b11, (short)0, d, false, false);

    // D rows 0..3 live in lanes 0..15, VGPRs 0..3: the 4 output channels of pixel (p, q).
    if (half == 0) {
      float* o = out + (size_t)plane * 65536u + (size_t)p * 256u + (size_t)((qc << 4) + m);
      __builtin_nontemporal_store(d[0], o);             // A_low0
      __builtin_nontemporal_store(d[1], o + 16777216);  // B_high1
      __builtin_nontemporal_store(d[2], o + 33554432);  // C_high2
      __builtin_nontemporal_store(d[3], o + 50331648);  // D_high3
    }
  }
}

extern "C" void kernel_launch(void* const* d_in, const int* in_sizes, int n_in,
                              void* d_out, int out_size, void* d_ws, size_t ws_size,
                              hipStream_t stream) {
  const float* x = (const float*)d_in[0];
  float* out = (float*)d_out;
  // 4096 blocks * 8 waves = 32768 waves; 1,048,576 chunks -> exactly 32 chunks/wave.
  nlwt_wmma_kernel<<<dim3(4096), dim3(256), 0, stream>>>(x, out);
}